// TopologyAwareBRepEncoder_90752658964794
// MI455X (gfx1250) — compile-verified
//
#include <hip/hip_runtime.h>
#include <math.h>

// ---------------- problem constants ----------------
constexpr int cB  = 16;
constexpr int cNF = 256;
constexpr int cNE = 512;
constexpr int cD  = 512;
constexpr int cDF = 48;
constexpr int cDE = 12;
constexpr int cNMSG = 3;
constexpr int cNTF  = 4;
constexpr int cNH = 8;
constexpr int cHD = 64;          // cD / cNH
constexpr int cS  = cNF + cNE;   // 768

typedef __attribute__((ext_vector_type(2))) float v2f;
typedef __attribute__((ext_vector_type(8))) float v8f;

// CDNA5 fp32 WMMA: D(16x16,f32) = A(16x4,f32) x B(4x16,f32) + C
// 8-arg form: (neg_a, A, neg_b, B, c_mod, C, reuse_a, reuse_b)
#define WMMA_F32(a, b, c) \
  __builtin_amdgcn_wmma_f32_16x16x4_f32(false, (a), false, (b), (short)0, (c), false, false)

__device__ __forceinline__ float gelu_exact(float x) {
  return 0.5f * x * (1.0f + erff(x * 0.70710678118654752f));
}
__device__ __forceinline__ float sigmoid_(float x) {
  return 1.0f / (1.0f + __expf(-x));
}
__device__ __forceinline__ float clipnn(float x, float lo, float hi) {
  if (x != x) return 0.0f;                 // NaN -> 0 (nan_to_num)
  return fminf(fmaxf(x, lo), hi);          // clamps +-inf too
}
__device__ __forceinline__ int clampi(int x, int lo, int hi) {
  return x < lo ? lo : (x > hi ? hi : x);
}

// =====================================================================
// Generic WMMA GEMM:  C[M,N] = act(A[M,K] @ W[K,N] + bias[N])
// block = 256 threads = 8 waves; C tile 32x64; wave grid 2x4 of 16x16 tiles
// act: 0=none, 1=gelu(exact), 2=sigmoid
// =====================================================================
__global__ void __launch_bounds__(256)
gemm_kernel(const float* __restrict__ A, const float* __restrict__ Wm,
            const float* __restrict__ bias, float* __restrict__ C,
            int M, int N, int K, int act)
{
  __shared__ float As[32][33];
  __shared__ float Bs[32][65];
  const int tid  = threadIdx.x;
  const int lane = tid & 31;
  const int wave = tid >> 5;
  const int half = lane >> 4;        // 0: lanes 0-15, 1: lanes 16-31
  const int l    = lane & 15;
  const int wr   = wave >> 2;        // 0..1
  const int wc   = wave & 3;         // 0..3
  const int mr   = wr * 16;
  const int nc   = wc * 16;
  const int rowBase = blockIdx.y * 32;
  const int colBase = blockIdx.x * 64;

  v8f acc = {};
  for (int k0 = 0; k0 < K; k0 += 32) {
    for (int i = tid; i < 32 * 32; i += 256) {
      int r = i >> 5, c = i & 31;
      int gr = rowBase + r, gc = k0 + c;
      As[r][c] = (gr < M && gc < K) ? A[(size_t)gr * K + gc] : 0.0f;
    }
    for (int i = tid; i < 32 * 64; i += 256) {
      int r = i >> 6, c = i & 63;
      int gr = k0 + r, gc = colBase + c;
      Bs[r][c] = (gr < K && gc < N) ? Wm[(size_t)gr * N + gc] : 0.0f;
    }
    __syncthreads();
#pragma unroll
    for (int kk = 0; kk < 32; kk += 4) {
      // A 16x4 frag: lanes<16 hold K=kk,kk+1; lanes>=16 hold K=kk+2,kk+3 (M=l)
      // B 4x16 frag: VGPR0 = row kk (/kk+2), VGPR1 = row kk+1 (/kk+3), N=l
      v2f a, b;
      a.x = As[mr + l][kk + 2 * half];
      a.y = As[mr + l][kk + 2 * half + 1];
      b.x = Bs[kk + 2 * half][nc + l];
      b.y = Bs[kk + 2 * half + 1][nc + l];
      acc = WMMA_F32(a, b, acc);
    }
    __syncthreads();
  }

  // C/D layout: VGPR v -> row (mr + half*8 + v), col (nc + l)
  int col = colBase + nc + l;
  if (col < N) {
    float bv = bias ? bias[col] : 0.0f;
#pragma unroll
    for (int v = 0; v < 8; ++v) {
      int row = rowBase + mr + half * 8 + v;
      if (row < M) {
        float x = acc[v] + bv;
        if (act == 1)      x = gelu_exact(x);
        else if (act == 2) x = sigmoid_(x);
        C[(size_t)row * N + col] = x;
      }
    }
  }
}

// =====================================================================
// Fused LayerNorm over D=512 rows (one block per row, 256 thr x 2 elems)
// y = x + add*gate*rowmask   (add/gate/rowmask optional)
// out = LN(y)*g + b ; flags&1 -> gelu(out) ; flags&2 -> nan_to_num(out)
// =====================================================================
__global__ void __launch_bounds__(256)
ln_kernel(const float* __restrict__ x, const float* __restrict__ add,
          const float* __restrict__ gate, const unsigned char* __restrict__ rowmask,
          const float* __restrict__ g, const float* __restrict__ bta,
          float* __restrict__ out, int flags)
{
  __shared__ float red[256];
  const int row = blockIdx.x;
  const int tid = threadIdx.x;
  const float rm = rowmask ? (rowmask[row] ? 1.0f : 0.0f) : 1.0f;

  float y[2];
  float s = 0.0f;
#pragma unroll
  for (int j = 0; j < 2; ++j) {
    int d = tid + j * 256;
    size_t i = (size_t)row * cD + d;
    float v = x[i];
    if (add) {
      float a = add[i];
      if (gate) a *= gate[i];
      v += a * rm;
    }
    y[j] = v; s += v;
  }
  red[tid] = s; __syncthreads();
  for (int o = 128; o > 0; o >>= 1) { if (tid < o) red[tid] += red[tid + o]; __syncthreads(); }
  float mean = red[0] * (1.0f / cD);
  __syncthreads();
  s = 0.0f;
#pragma unroll
  for (int j = 0; j < 2; ++j) { float d = y[j] - mean; s += d * d; }
  red[tid] = s; __syncthreads();
  for (int o = 128; o > 0; o >>= 1) { if (tid < o) red[tid] += red[tid + o]; __syncthreads(); }
  float rstd = rsqrtf(red[0] * (1.0f / cD) + 1e-5f);
#pragma unroll
  for (int j = 0; j < 2; ++j) {
    int d = tid + j * 256;
    float v = (y[j] - mean) * rstd * g[d] + bta[d];
    if (flags & 1) v = gelu_exact(v);
    if (flags & 2) {
      if (v != v) v = 0.0f;
      v = fminf(fmaxf(v, -3.4028234663852886e38f), 3.4028234663852886e38f);
    }
    out[(size_t)row * cD + d] = v;
  }
}

// =====================================================================
// Flash-style fused attention, one block per (b, h, 64-query tile)
// 256 threads = 8 waves in a 4x2 grid; each wave owns a 16x32 output slab
// =====================================================================
__global__ void __launch_bounds__(256)
attention_kernel(const float* __restrict__ qkv,
                 const unsigned char* __restrict__ fmask,
                 const unsigned char* __restrict__ emask,
                 float* __restrict__ out)
{
  __shared__ float Qs[64 * 64];
  __shared__ float Ks[64 * 64];       // reused for K then V
  __shared__ float Ss[64 * 64];       // score / prob tile
  __shared__ float rowM[64], rowL[64], rowC[64];

  const int b = blockIdx.z, h = blockIdx.y, qt = blockIdx.x;
  const int tid  = threadIdx.x;
  const int lane = tid & 31;
  const int wave = tid >> 5;
  const int half = lane >> 4;
  const int l    = lane & 15;
  const int wr   = wave >> 1;         // 0..3 : query-row group
  const int wc   = wave & 1;          // 0..1 : col group (32 wide)
  const int mb   = wr * 16;
  const float scale = 0.125f;         // 1/sqrt(64)
  const int QKVW = 3 * cD;

  for (int i = tid; i < 64; i += 256) { rowM[i] = -1e30f; rowL[i] = 0.0f; }
  for (int i = tid; i < 64 * 64; i += 256) {
    int r = i >> 6, d = i & 63;
    Qs[i] = qkv[((size_t)b * cS + qt * 64 + r) * QKVW + h * cHD + d];
  }
  __syncthreads();

  v8f o0 = {}, o1 = {};

  for (int kt = 0; kt < cS / 64; ++kt) {
    // ---- stage K tile (64 keys x 64 hd) ----
    for (int i = tid; i < 64 * 64; i += 256) {
      int r = i >> 6, d = i & 63;
      Ks[i] = qkv[((size_t)b * cS + kt * 64 + r) * QKVW + cD + h * cHD + d];
    }
    __syncthreads();

    // ---- S = Q @ K^T  (each wave: two 16x16 tiles) ----
#pragma unroll
    for (int t = 0; t < 2; ++t) {
      v8f acc = {};
      int nb = wc * 32 + t * 16;
#pragma unroll
      for (int ko = 0; ko < cHD; ko += 4) {
        v2f a, bb;
        a.x  = Qs[(mb + l) * 64 + ko + 2 * half];
        a.y  = Qs[(mb + l) * 64 + ko + 2 * half + 1];
        bb.x = Ks[(nb + l) * 64 + ko + 2 * half];   // B[kk][n] = K[n][kk]
        bb.y = Ks[(nb + l) * 64 + ko + 2 * half + 1];
        acc = WMMA_F32(a, bb, acc);
      }
      int colk = nb + l;
      int sk = kt * 64 + colk;
      bool mk = (sk < cNF) ? (fmask[b * cNF + sk] != 0)
                           : (emask[b * cNE + sk - cNF] != 0);
#pragma unroll
      for (int v = 0; v < 8; ++v) {
        int row = mb + half * 8 + v;
        Ss[row * 64 + colk] = mk ? acc[v] * scale : -1e9f;
      }
    }
    __syncthreads();

    // ---- online softmax over this key tile (wc==0 waves, lanes 0..15) ----
    if (wc == 0 && lane < 16) {
      int r = mb + lane;
      float mOld = rowM[r];
      float mNew = mOld;
      for (int c = 0; c < 64; ++c) mNew = fmaxf(mNew, Ss[r * 64 + c]);
      float corr = __expf(mOld - mNew);
      float sum = 0.0f;
      for (int c = 0; c < 64; ++c) {
        float e = __expf(Ss[r * 64 + c] - mNew);
        Ss[r * 64 + c] = e;
        sum += e;
      }
      rowL[r] = rowL[r] * corr + sum;
      rowM[r] = mNew;
      rowC[r] = corr;
    }
    __syncthreads();

    // rescale running O by correction
#pragma unroll
    for (int v = 0; v < 8; ++v) {
      float c = rowC[mb + half * 8 + v];
      o0[v] *= c; o1[v] *= c;
    }

    // ---- stage V tile into Ks ----
    for (int i = tid; i < 64 * 64; i += 256) {
      int r = i >> 6, d = i & 63;
      Ks[i] = qkv[((size_t)b * cS + kt * 64 + r) * QKVW + 2 * cD + h * cHD + d];
    }
    __syncthreads();

    // ---- O += P @ V ----
#pragma unroll
    for (int ko = 0; ko < 64; ko += 4) {
      v2f a, b0, b1;
      a.x = Ss[(mb + l) * 64 + ko + 2 * half];
      a.y = Ss[(mb + l) * 64 + ko + 2 * half + 1];
      int n0 = wc * 32 + l;
      b0.x = Ks[(ko + 2 * half) * 64 + n0];
      b0.y = Ks[(ko + 2 * half + 1) * 64 + n0];
      b1.x = Ks[(ko + 2 * half) * 64 + n0 + 16];
      b1.y = Ks[(ko + 2 * half + 1) * 64 + n0 + 16];
      o0 = WMMA_F32(a, b0, o0);
      o1 = WMMA_F32(a, b1, o1);
    }
    __syncthreads();
  }

  // ---- finalize and write O (B,S,D) with head-major packing ----
#pragma unroll
  for (int v = 0; v < 8; ++v) {
    int row = mb + half * 8 + v;
    float inv = 1.0f / rowL[row];
    int s = qt * 64 + row;
    size_t base = ((size_t)b * cS + s) * cD + h * cHD;
    out[base + wc * 32 + l]      = o0[v] * inv;
    out[base + wc * 32 + 16 + l] = o1[v] * inv;
  }
}

// =====================================================================
// Elementwise kernels
// =====================================================================
__global__ void face_embed_kernel(float* __restrict__ Fh,
    const float* __restrict__ ft, const float* __restrict__ cw, const float* __restrict__ cb,
    const float* __restrict__ nw, const float* __restrict__ nb2,
    const float* __restrict__ aw, const float* __restrict__ ab,
    const float* __restrict__ lemb,
    const float* __restrict__ fc, const float* __restrict__ fn,
    const float* __restrict__ fa, const int* __restrict__ bfs)
{
  int idx = blockIdx.x * 256 + threadIdx.x;
  if (idx >= cB * cNF * cD) return;
  int d = idx & (cD - 1);
  int r = idx >> 9;
  float c0 = clipnn(fc[r * 3 + 0], -1e4f, 1e4f);
  float c1 = clipnn(fc[r * 3 + 1], -1e4f, 1e4f);
  float c2 = clipnn(fc[r * 3 + 2], -1e4f, 1e4f);
  float n0 = clipnn(fn[r * 3 + 0], -1.0f, 1.0f);
  float n1 = clipnn(fn[r * 3 + 1], -1.0f, 1.0f);
  float n2 = clipnn(fn[r * 3 + 2], -1.0f, 1.0f);
  float a  = clipnn(fa[r], 0.0f, 1e6f);
  int lv = clampi(bfs[r], 0, 31);
  Fh[idx] += ft[d]
    + c0 * cw[d] + c1 * cw[cD + d] + c2 * cw[2 * cD + d] + cb[d]
    + n0 * nw[d] + n1 * nw[cD + d] + n2 * nw[2 * cD + d] + nb2[d]
    + a * aw[d] + ab[d]
    + lemb[(size_t)lv * cD + d];
}

__global__ void edge_embed_kernel(float* __restrict__ Eh,
    const float* __restrict__ et,
    const float* __restrict__ mw, const float* __restrict__ mb2,
    const float* __restrict__ lw, const float* __restrict__ lb,
    const float* __restrict__ em, const float* __restrict__ el)
{
  int idx = blockIdx.x * 256 + threadIdx.x;
  if (idx >= cB * cNE * cD) return;
  int d = idx & (cD - 1);
  int r = idx >> 9;
  float m0 = clipnn(em[r * 3 + 0], -1e4f, 1e4f);
  float m1 = clipnn(em[r * 3 + 1], -1e4f, 1e4f);
  float m2 = clipnn(em[r * 3 + 2], -1e4f, 1e4f);
  float L  = clipnn(el[r], 0.0f, 1e6f);
  Eh[idx] += et[d]
    + m0 * mw[d] + m1 * mw[cD + d] + m2 * mw[2 * cD + d] + mb2[d]
    + L * lw[d] + lb[d];
}

__global__ void valid_kernel(float* __restrict__ valid,
    const unsigned char* __restrict__ emask, const int* __restrict__ e2f)
{
  int r = blockIdx.x * 256 + threadIdx.x;
  if (r >= cB * cNE) return;
  valid[r] = (emask[r] && e2f[2 * r] >= 0 && e2f[2 * r + 1] >= 0) ? 1.0f : 0.0f;
}

__global__ void concat3_kernel(float* __restrict__ out, const float* __restrict__ Eh,
                               const float* __restrict__ Fh, const int* __restrict__ e2f)
{
  const int W3 = 3 * cD;
  int idx = blockIdx.x * 256 + threadIdx.x;
  if (idx >= cB * cNE * W3) return;
  int c = idx % W3;
  int r = idx / W3;
  int bb = r / cNE;
  float v;
  if (c < cD) {
    v = Eh[(size_t)r * cD + c];
  } else {
    int which = (c < 2 * cD) ? 0 : 1;
    int f = clampi(e2f[r * 2 + which], 0, cNF - 1);
    v = Fh[((size_t)bb * cNF + f) * cD + (c - cD - which * cD)];
  }
  out[idx] = v;
}

__global__ void concat2_kernel(float* __restrict__ out, const float* __restrict__ X0,
                               const float* __restrict__ X1, int rows)
{
  const int W2 = 2 * cD;
  int idx = blockIdx.x * 256 + threadIdx.x;
  if (idx >= rows * W2) return;
  int c = idx % W2;
  int r = idx / W2;
  out[idx] = (c < cD) ? X0[(size_t)r * cD + c] : X1[(size_t)r * cD + c - cD];
}

__global__ void rowscale_kernel(float* __restrict__ x, const float* __restrict__ s, int rows)
{
  int idx = blockIdx.x * 256 + threadIdx.x;
  if (idx >= rows * cD) return;
  x[idx] *= s[idx >> 9];
}

__global__ void fill_kernel(float* __restrict__ p, float v, int n)
{
  int idx = blockIdx.x * 256 + threadIdx.x;
  if (idx < n) p[idx] = v;
}

__global__ void scatter_kernel(float* __restrict__ fm, float* __restrict__ cnt,
                               const float* __restrict__ Eh, const float* __restrict__ valid,
                               const int* __restrict__ e2f)
{
  int idx = blockIdx.x * 256 + threadIdx.x;
  if (idx >= cB * cNE * cD) return;
  int d = idx & (cD - 1);
  int r = idx >> 9;
  float vl = valid[r];
  float v = Eh[idx] * vl;
  int bb = r / cNE;
  int f1 = clampi(e2f[2 * r], 0, cNF - 1);
  int f2 = clampi(e2f[2 * r + 1], 0, cNF - 1);
  atomicAdd(&fm[((size_t)bb * cNF + f1) * cD + d], v);
  atomicAdd(&fm[((size_t)bb * cNF + f2) * cD + d], v);
  if (d == 0) {
    atomicAdd(&cnt[bb * cNF + f1], vl);
    atomicAdd(&cnt[bb * cNF + f2], vl);
  }
}

__global__ void divcnt_kernel(float* __restrict__ fm, const float* __restrict__ cnt)
{
  int idx = blockIdx.x * 256 + threadIdx.x;
  if (idx >= cB * cNF * cD) return;
  fm[idx] /= (cnt[idx >> 9] + 1e-8f);
}

__global__ void buildx_kernel(float* __restrict__ X, const float* __restrict__ Fh,
                              const float* __restrict__ Eh)
{
  int idx = blockIdx.x * 256 + threadIdx.x;
  if (idx >= cB * cS * cD) return;
  int d = idx & (cD - 1);
  int rs = idx >> 9;
  int s = rs % cS;
  int bb = rs / cS;
  X[idx] = (s < cNF) ? Fh[((size_t)bb * cNF + s) * cD + d]
                     : Eh[((size_t)bb * cNE + (s - cNF)) * cD + d];
}

__global__ void faces_out_kernel(float* __restrict__ out2, const float* __restrict__ Xall)
{
  int idx = blockIdx.x * 256 + threadIdx.x;
  if (idx >= cB * cNF * cD) return;
  int d = idx & (cD - 1);
  int r = idx >> 9;
  int s = r % cNF;
  int bb = r / cNF;
  out2[idx] = Xall[((size_t)bb * cS + s) * cD + d];
}

// =====================================================================
// Host orchestration
// =====================================================================
// param leaf indices (dict insertion order of setup_inputs()['params'])
enum {
  I_FP_W0 = 0, I_FP_B0, I_FP_G, I_FP_BN, I_FP_W1, I_FP_B1,
  I_EP_W0, I_EP_B0, I_EP_G, I_EP_BN, I_EP_W1, I_EP_B1,
  I_CEN_W, I_CEN_B, I_NRM_W, I_NRM_B, I_AREA_W, I_AREA_B,
  I_EMID_W, I_EMID_B, I_ELEN_W, I_ELEN_B,
  I_LEVEL, I_FTYPE, I_ETYPE,
  I_MSG0 = 25,   // + l*16 : f2e_w0,f2e_b0,f2e_w1,f2e_b1,e2f_w0,e2f_b0,e2f_w1,e2f_b1,
                 //          ge_w,ge_b,gf_w,gf_b,lnf_g,lnf_b,lne_g,lne_b
  I_TF0 = 73,    // + l*12 : qkv_w,qkv_b,out_w,out_b,ln1_g,ln1_b,ln2_g,ln2_b,
                 //          ff_w0,ff_b0,ff_w1,ff_b1
  I_FN_G = 121, I_FN_B = 122
};

static inline int ceildiv(int a, int b) { return (a + b - 1) / b; }

extern "C" void kernel_launch(void* const* d_in, const int* in_sizes, int n_in,
                              void* d_out, int out_size, void* d_ws, size_t ws_size,
                              hipStream_t stream)
{
  (void)in_sizes; (void)out_size;
  if (n_in < 134) return;
  const int p0 = n_in - 11;   // first non-param input (params are leaves 0..p0-1)

  auto PRM = [&](int i) -> const float* { return (const float*)d_in[i]; };

  const float*         face_feats = (const float*)d_in[p0 + 0];
  const float*         edge_feats = (const float*)d_in[p0 + 1];
  const unsigned char* fmask      = (const unsigned char*)d_in[p0 + 2];
  const unsigned char* emask      = (const unsigned char*)d_in[p0 + 3];
  const int*           e2f        = (const int*)d_in[p0 + 4];
  const float*         fc         = (const float*)d_in[p0 + 5];
  const float*         fnrm       = (const float*)d_in[p0 + 6];
  const float*         fa         = (const float*)d_in[p0 + 7];
  const int*           bfs        = (const int*)d_in[p0 + 8];
  const float*         emid       = (const float*)d_in[p0 + 9];
  const float*         elen       = (const float*)d_in[p0 + 10];

  const int MF = cB * cNF;   // 4096
  const int ME = cB * cNE;   // 8192
  const int MS = cB * cS;    // 12288

  // ---- workspace carve (floats) ----
  float* ws = (float*)d_ws;
  size_t off = 0;
  auto carve = [&](size_t n) { float* p = ws + off; off += n; return p; };
  float* Fh    = carve((size_t)MF * cD);
  float* Eh    = carve((size_t)ME * cD);
  float* Xb    = carve((size_t)MS * cD);
  float* BUF1  = carve((size_t)MS * 3 * cD);   // concat inputs / qkv
  float* BUF2  = carve((size_t)MS * 4 * cD);   // MLP hidden / attention out
  float* BUF3  = carve((size_t)MS * cD);       // emsg / fup / T1
  float* BUF4  = carve((size_t)MS * cD);       // gates
  float* FM    = carve((size_t)MF * cD);
  float* CNT   = carve((size_t)MF);
  float* VALID = carve((size_t)ME);
  if (off * sizeof(float) > ws_size) return;

  auto gemm = [&](const float* A, const float* Wm, const float* bias, float* C,
                  int M, int N, int K, int act) {
    dim3 grid(ceildiv(N, 64), ceildiv(M, 32));
    gemm_kernel<<<grid, 256, 0, stream>>>(A, Wm, bias, C, M, N, K, act);
  };
  auto ln = [&](const float* x, const float* add, const float* gate,
                const unsigned char* rmask, const float* g, const float* b2,
                float* out, int rows, int flags) {
    ln_kernel<<<rows, 256, 0, stream>>>(x, add, gate, rmask, g, b2, out, flags);
  };
  auto ew = [&](int n) { return ceildiv(n, 256); };

  // ================= face / edge input projections =================
  gemm(face_feats, PRM(I_FP_W0), PRM(I_FP_B0), BUF2, MF, cD, cDF, 0);
  ln(BUF2, nullptr, nullptr, nullptr, PRM(I_FP_G), PRM(I_FP_BN), BUF3, MF, 1);
  gemm(BUF3, PRM(I_FP_W1), PRM(I_FP_B1), Fh, MF, cD, cD, 0);
  face_embed_kernel<<<ew(MF * cD), 256, 0, stream>>>(
      Fh, PRM(I_FTYPE), PRM(I_CEN_W), PRM(I_CEN_B), PRM(I_NRM_W), PRM(I_NRM_B),
      PRM(I_AREA_W), PRM(I_AREA_B), PRM(I_LEVEL), fc, fnrm, fa, bfs);

  gemm(edge_feats, PRM(I_EP_W0), PRM(I_EP_B0), BUF2, ME, cD, cDE, 0);
  ln(BUF2, nullptr, nullptr, nullptr, PRM(I_EP_G), PRM(I_EP_BN), BUF3, ME, 1);
  gemm(BUF3, PRM(I_EP_W1), PRM(I_EP_B1), Eh, ME, cD, cD, 0);
  edge_embed_kernel<<<ew(ME * cD), 256, 0, stream>>>(
      Eh, PRM(I_ETYPE), PRM(I_EMID_W), PRM(I_EMID_B), PRM(I_ELEN_W), PRM(I_ELEN_B),
      emid, elen);

  valid_kernel<<<ew(ME), 256, 0, stream>>>(VALID, emask, e2f);

  // ================= message passing =================
  for (int lyr = 0; lyr < cNMSG; ++lyr) {
    const int pm = I_MSG0 + lyr * 16;
    // edges: emsg = lin(gelu(lin(cat(Eh, Fh[f1], Fh[f2])))) * valid
    concat3_kernel<<<ew(ME * 3 * cD), 256, 0, stream>>>(BUF1, Eh, Fh, e2f);
    gemm(BUF1, PRM(pm + 0), PRM(pm + 1), BUF2, ME, 2 * cD, 3 * cD, 1);
    gemm(BUF2, PRM(pm + 2), PRM(pm + 3), BUF3, ME, cD, 2 * cD, 0);
    rowscale_kernel<<<ew(ME * cD), 256, 0, stream>>>(BUF3, VALID, ME);
    // gate + gated residual LN
    concat2_kernel<<<ew(ME * 2 * cD), 256, 0, stream>>>(BUF1, Eh, BUF3, ME);
    gemm(BUF1, PRM(pm + 8), PRM(pm + 9), BUF4, ME, cD, 2 * cD, 2);
    ln(Eh, BUF3, BUF4, nullptr, PRM(pm + 14), PRM(pm + 15), Eh, ME, 0);
    // scatter-mean edges -> faces
    fill_kernel<<<ew(MF * cD), 256, 0, stream>>>(FM, 0.0f, MF * cD);
    fill_kernel<<<ew(MF), 256, 0, stream>>>(CNT, 0.0f, MF);
    scatter_kernel<<<ew(ME * cD), 256, 0, stream>>>(FM, CNT, Eh, VALID, e2f);
    divcnt_kernel<<<ew(MF * cD), 256, 0, stream>>>(FM, CNT);
    // faces: fup = lin(gelu(lin(cat(Fh, fmsg))))
    concat2_kernel<<<ew(MF * 2 * cD), 256, 0, stream>>>(BUF1, Fh, FM, MF);
    gemm(BUF1, PRM(pm + 4), PRM(pm + 5), BUF2, MF, cD, 2 * cD, 1);
    gemm(BUF2, PRM(pm + 6), PRM(pm + 7), BUF3, MF, cD, cD, 0);
    concat2_kernel<<<ew(MF * 2 * cD), 256, 0, stream>>>(BUF1, Fh, BUF3, MF);
    gemm(BUF1, PRM(pm + 10), PRM(pm + 11), BUF4, MF, cD, 2 * cD, 2);
    ln(Fh, BUF3, BUF4, fmask, PRM(pm + 12), PRM(pm + 13), Fh, MF, 0);
  }

  // ================= transformer stack =================
  buildx_kernel<<<ew(MS * cD), 256, 0, stream>>>(Xb, Fh, Eh);
  for (int lyr = 0; lyr < cNTF; ++lyr) {
    const int pt = I_TF0 + lyr * 12;
    gemm(Xb, PRM(pt + 0), PRM(pt + 1), BUF1, MS, 3 * cD, cD, 0);   // qkv
    attention_kernel<<<dim3(cS / 64, cNH, cB), 256, 0, stream>>>(BUF1, fmask, emask, BUF2);
    gemm(BUF2, PRM(pt + 2), PRM(pt + 3), BUF3, MS, cD, cD, 0);     // out proj
    ln(Xb, BUF3, nullptr, nullptr, PRM(pt + 4), PRM(pt + 5), Xb, MS, 0);
    gemm(Xb, PRM(pt + 8), PRM(pt + 9), BUF2, MS, 4 * cD, cD, 1);   // ff0 + gelu
    gemm(BUF2, PRM(pt + 10), PRM(pt + 11), BUF3, MS, cD, 4 * cD, 0);
    ln(Xb, BUF3, nullptr, nullptr, PRM(pt + 6), PRM(pt + 7), Xb, MS, 0);
  }

  // ================= final norm + outputs =================
  float* out1 = (float*)d_out;                 // (B,S,D)
  float* out2 = out1 + (size_t)MS * cD;        // (B,NF,D)
  ln(Xb, nullptr, nullptr, nullptr, PRM(I_FN_G), PRM(I_FN_B), out1, MS, 2);
  faces_out_kernel<<<ew(MF * cD), 256, 0, stream>>>(out2, out1);
}